// NonLocalMeansDenoising_56573309223712
// MI455X (gfx1250) — compile-verified
//
#include <hip/hip_runtime.h>

// ---------------------------------------------------------------------------
// NLM denoise, fused single kernel.
// Box sums (21x21) computed as banded-matrix matmuls on the matrix pipe via
// V_WMMA_F32_16X16X4_F32 (exact f32). Batch reduction in registers.
// Output broadcast stores are non-temporal (write-once stream, 25 MB) so the
// streamed result does not evict input halo lines from L2.
// ---------------------------------------------------------------------------

typedef float v2f __attribute__((ext_vector_type(2)));
typedef float v4f __attribute__((ext_vector_type(4)));
typedef float v8f __attribute__((ext_vector_type(8)));

#define WINDOW 21
#define PAD    10
#define NWIN   441.0f
#define H2INV  100.0f      // 1 / (0.1*0.1)
#define EPSV   1e-10f

#define TX 64              // tile width  (output)
#define TY 32              // tile height (output)
#define PXD 84             // padded cols used  = TX + 2*PAD
#define PYD 52             // padded rows used  = TY + 2*PAD
#define PSTR 88            // LDS stride of P (floats)
#define PROWS 64           // alloc rows for P (ceil16 of 52; rows 52..63 junk, never consumed)
#define HSTR 68            // LDS stride of H (floats)
#define HROWS 64           // H rows 0..51 used (TY+20), groups of 16 -> 64
#define SSTR 68            // LDS stride of S

__device__ __forceinline__ int reflect512(int i) {
    i = (i < 0) ? -i : i;
    i = (i > 511) ? (1022 - i) : i;
    return i;
}

// Banded 0/1 fragment, value[k][n] = 1.0 iff 0 <= (i0 + k) - n <= 20.
// Layout works as both A(16x4, M=lane&15) and B(4x16, N=lane&15):
//   VGPR0 holds K = 2*(lane>>4), VGPR1 holds K+1 (per ISA 7.12.2).
__device__ __forceinline__ v2f band_frag(int i0, int lane) {
    int n  = lane & 15;
    int k0 = i0 + 2 * (lane >> 4);
    int d0 = k0 - n;
    int d1 = d0 + 1;
    v2f f;
    f.x = (d0 >= 0 && d0 <= 20) ? 1.0f : 0.0f;
    f.y = (d1 >= 0 && d1 <= 20) ? 1.0f : 0.0f;
    return f;
}

__global__ __launch_bounds__(256)
void nlm_fused_kernel(const float* __restrict__ noisy, float* __restrict__ out) {
    __shared__ float P [PROWS * PSTR];   // padded tile (reflect), rows 0..51 valid
    __shared__ float H1[HROWS * HSTR];   // horizontal box sums of P
    __shared__ float H2[HROWS * HSTR];   // horizontal box sums of P^2
    __shared__ float S1[TY   * SSTR];    // full 21x21 sums
    __shared__ float S2[TY   * SSTR];

    const int tid  = threadIdx.x;
    const int lane = tid & 31;
    const int wave = tid >> 5;

    const int tx = blockIdx.x;          // 0..7
    const int ty = blockIdx.y;          // 0..15
    const int c  = blockIdx.z;          // 0..2
    const int x0 = tx * TX;
    const int y0 = ty * TY;

    // WMMA lane decomposition (shared by all fragment types here)
    const int lm = lane & 15;           // M (A rows) / N (B,D cols)
    const int kp = (lane >> 4) * 2;     // K pair base within 4-chunk

    // Precompute the 9 band fragments (K blocks i0 = 0,4,...,32)
    v2f band[9];
#pragma unroll
    for (int kb = 0; kb < 9; ++kb) band[kb] = band_frag(kb * 4, lane);

    // pixel-phase ownership: 2 rows x 4 cols per thread
    const int prow0 = (tid >> 4) * 2;   // 0..30
    const int pcol0 = (tid & 15) * 4;   // 0..60

    float num[2][4], den[2][4];
#pragma unroll
    for (int i = 0; i < 2; ++i)
#pragma unroll
        for (int j = 0; j < 4; ++j) { num[i][j] = 0.0f; den[i][j] = 0.0f; }

#pragma unroll 1
    for (int b = 0; b < 8; ++b) {
        const float* plane = noisy + (size_t)(b * 3 + c) * (512 * 512);

        // ---- load padded tile with reflect indexing ----
        for (int idx = tid; idx < PYD * PXD; idx += 256) {
            int py = idx / PXD;
            int px = idx - py * PXD;
            int gy = reflect512(y0 + py - PAD);
            int gx = reflect512(x0 + px - PAD);
            P[py * PSTR + px] = plane[gy * 512 + gx];
        }
        __syncthreads();

        // ---- horizontal pass: H[y][n] = sum_{k=0..20} P[y][n+k]  (and P^2) ----
        // 16 tile-ops (4 ygroups x 4 xgroups), 2 per wave. data-as-A, band-as-B.
#pragma unroll
        for (int j = 0; j < 2; ++j) {
            int t = wave + j * 8;
            int ybase = (t >> 2) * 16;
            int xbase = (t & 3) * 16;
            v8f acc1 = {}; v8f acc2 = {};
#pragma unroll
            for (int kb = 0; kb < 9; ++kb) {
                const float* rp = &P[(ybase + lm) * PSTR + xbase + kb * 4 + kp];
                v2f a;  a.x  = rp[0];        a.y  = rp[1];
                v2f a2; a2.x = a.x * a.x;    a2.y = a.y * a.y;
                acc1 = __builtin_amdgcn_wmma_f32_16x16x4_f32(
                           false, a,  false, band[kb], (short)0, acc1, false, false);
                acc2 = __builtin_amdgcn_wmma_f32_16x16x4_f32(
                           false, a2, false, band[kb], (short)0, acc2, false, false);
            }
            int col   = xbase + lm;
            int rbase = ybase + 8 * (lane >> 4);
#pragma unroll
            for (int r = 0; r < 8; ++r) {
                H1[(rbase + r) * HSTR + col] = acc1[r];
                H2[(rbase + r) * HSTR + col] = acc2[r];
            }
        }
        __syncthreads();

        // ---- vertical pass: S[m][n] = sum_{k=0..20} H[m+k][n] ----
        // 8 tile-ops (2 ygroups x 4 xgroups), 1 per wave. band-as-A, data-as-B.
        {
            int ybase = (wave >> 2) * 16;
            int xbase = (wave & 3) * 16;
            int col   = xbase + lm;
            v8f acc1 = {}; v8f acc2 = {};
#pragma unroll
            for (int kb = 0; kb < 9; ++kb) {
                int row = ybase + kb * 4 + kp;
                v2f b1; b1.x = H1[row * HSTR + col]; b1.y = H1[(row + 1) * HSTR + col];
                v2f b2; b2.x = H2[row * HSTR + col]; b2.y = H2[(row + 1) * HSTR + col];
                acc1 = __builtin_amdgcn_wmma_f32_16x16x4_f32(
                           false, band[kb], false, b1, (short)0, acc1, false, false);
                acc2 = __builtin_amdgcn_wmma_f32_16x16x4_f32(
                           false, band[kb], false, b2, (short)0, acc2, false, false);
            }
            int rbase = ybase + 8 * (lane >> 4);
#pragma unroll
            for (int r = 0; r < 8; ++r) {
                S1[(rbase + r) * SSTR + col] = acc1[r];
                S2[(rbase + r) * SSTR + col] = acc2[r];
            }
        }
        __syncthreads();

        // ---- per-pixel weights + batch accumulation ----
#pragma unroll
        for (int i = 0; i < 2; ++i) {
            int m = prow0 + i;
#pragma unroll
            for (int jx = 0; jx < 4; ++jx) {
                int n   = pcol0 + jx;
                float x  = P[(m + PAD) * PSTR + (n + PAD)];  // center pixel
                float s1 = S1[m * SSTR + n];
                float s2 = S2[m * SSTR + n];
                float d  = (s2 - 2.0f * x * s1 + NWIN * x * x) * (1.0f / NWIN);
                float w  = __expf(-d * H2INV);
                num[i][jx] += w * P[m * PSTR + n];           // patch = padded[:H,:W]
                den[i][jx] += w;
            }
        }
        __syncthreads();   // P/H/S reused next batch iteration
    }

    // ---- normalize, clip, broadcast to all 8 batch slices (non-temporal) ----
#pragma unroll
    for (int i = 0; i < 2; ++i) {
        int gy = y0 + prow0 + i;
        v4f o;
#pragma unroll
        for (int jx = 0; jx < 4; ++jx) {
            float v = num[i][jx] / (den[i][jx] + EPSV);
            o[jx] = fminf(fmaxf(v, 0.0f), 1.0f);
        }
        size_t base = ((size_t)c * 512 + gy) * 512 + (x0 + pcol0);
#pragma unroll
        for (int b = 0; b < 8; ++b) {
            __builtin_nontemporal_store(o, (v4f*)(out + base + (size_t)b * 3 * 512 * 512));
        }
    }
}

extern "C" void kernel_launch(void* const* d_in, const int* in_sizes, int n_in,
                              void* d_out, int out_size, void* d_ws, size_t ws_size,
                              hipStream_t stream) {
    (void)in_sizes; (void)n_in; (void)out_size; (void)d_ws; (void)ws_size;
    const float* noisy = (const float*)d_in[0];
    float*       out   = (float*)d_out;
    dim3 grid(512 / TX, 512 / TY, 3);   // 8 x 16 x 3 = 384 workgroups
    nlm_fused_kernel<<<grid, 256, 0, stream>>>(noisy, out);
}